// GAT_base_47614007443723
// MI455X (gfx1250) — compile-verified
//
#include <hip/hip_runtime.h>
#include <hip/hip_bf16.h>

// ---------------------------------------------------------------------------
// GATv2, 3 layers, N=50000 nodes, E=800000 edges (+N self loops), F=HID=256,
// HEADS=4, C=64, edge_dim=64.
//
// MI455X reasoning: xl/xr/out node matrices are 51MB each -> all L2-resident
// (192MB), so per-edge gathers & fp32 atomic scatters run at L2 speed.
// Matmuls (~123 GFLOP total) go through V_WMMA_F32_16X16X4_F32 to keep fp32
// reference precision; the edge transform (eattr @ We) is fused into the
// logits kernel as a 16-edge-per-wave WMMA so the 870MB "ee" tensor is never
// materialized.
// ---------------------------------------------------------------------------

typedef __attribute__((ext_vector_type(2))) float v2f;
typedef __attribute__((ext_vector_type(8))) float v8f;

constexpr int   kN      = 50000;
constexpr int   kE      = 800000;
constexpr int   kEf     = kE + kN;        // 850000, divisible by 16
constexpr int   kF      = 256;            // F_IN == HID == OUT
constexpr int   kED     = 64;             // edge dim
constexpr int   kHeads  = 4;
constexpr int   kLayers = 3;
constexpr int   kGroups = kEf / 16;       // 53125 edge groups of 16

// ---- ordered-uint encoding so atomicMax(uint) == float max ------------------
__device__ __forceinline__ unsigned f2ord(float f) {
  unsigned u = __float_as_uint(f);
  return (u & 0x80000000u) ? ~u : (u | 0x80000000u);
}
__device__ __forceinline__ float ord2f(unsigned s) {
  unsigned u = (s & 0x80000000u) ? (s & 0x7FFFFFFFu) : ~s;
  return __uint_as_float(u);
}
constexpr unsigned kOrdNegInf = 0x007FFFFFu;  // f2ord(-inf)

__device__ __forceinline__ v8f wmma_f32_4(v2f a, v2f b, v8f c) {
  // D = A(16x4 f32) * B(4x16 f32) + C(16x16 f32)
  return __builtin_amdgcn_wmma_f32_16x16x4_f32(false, a, false, b,
                                               (short)0, c, false, false);
}

// ---------------------------------------------------------------------------
// Preprocess: loop_attr[n] = mean over edges with dst==n of edge_attr
// ---------------------------------------------------------------------------
__global__ void pre_sum_kernel(const float* __restrict__ eattr,
                               const long long* __restrict__ ei,
                               float* __restrict__ loopSum,
                               float* __restrict__ cnt) {
  int idx = blockIdx.x * 256 + threadIdx.x;
  if (idx >= kE * kED) return;
  int e = idx >> 6;
  int k = idx & 63;
  int d = (int)ei[kE + e];
  atomicAdd(&loopSum[(size_t)d * kED + k], eattr[idx]);
  if (k == 0) atomicAdd(&cnt[d], 1.0f);
}

__global__ void pre_div_kernel(float* __restrict__ loopAttr,
                               const float* __restrict__ cnt) {
  int idx = blockIdx.x * 256 + threadIdx.x;
  if (idx >= kN * kED) return;
  loopAttr[idx] = loopAttr[idx] / fmaxf(cnt[idx >> 6], 1.0f);
}

// ---------------------------------------------------------------------------
// Per-layer inits
// ---------------------------------------------------------------------------
__global__ void init_nh_kernel(unsigned* __restrict__ lmaxU,
                               float* __restrict__ denom) {
  int idx = blockIdx.x * 256 + threadIdx.x;
  if (idx >= kN * kHeads) return;
  lmaxU[idx] = kOrdNegInf;
  denom[idx] = 0.0f;
}

__global__ void init_out_kernel(float* __restrict__ xout,
                                const float* __restrict__ bias) {
  int idx = blockIdx.x * 256 + threadIdx.x;
  if (idx >= kN * kF) return;
  xout[idx] = bias[idx & (kF - 1)];
}

// ---------------------------------------------------------------------------
// Node GEMM: Y = X @ W + b   for (Wl,bl)->xl and (Wr,br)->xr  (blockIdx.z)
// Block = 256 thr (8 waves). Each wave: one 16x16 f32 WMMA tile, K=256.
// A tile (16x256) staged in LDS; W (256KB) served from L2.
// ---------------------------------------------------------------------------
__global__ void node_gemm_kernel(const float* __restrict__ X,
                                 const float* __restrict__ W0,
                                 const float* __restrict__ b0,
                                 const float* __restrict__ W1,
                                 const float* __restrict__ b1,
                                 float* __restrict__ Y0,
                                 float* __restrict__ Y1) {
  __shared__ float sA[16 * 260];  // padded stride vs 64-bank LDS

  const float* W    = blockIdx.z ? W1 : W0;
  const float* bias = blockIdx.z ? b1 : b0;
  float*       Y    = blockIdx.z ? Y1 : Y0;

  const int row0 = blockIdx.x * 16;
  const int tid  = threadIdx.x;

#pragma unroll
  for (int j = 0; j < 16; ++j)
    sA[j * 260 + tid] = X[(size_t)(row0 + j) * kF + tid];
  __syncthreads();

  const int wave = tid >> 5;
  const int lane = tid & 31;
  const int half = lane >> 4;   // 0: K+0/+1 side, 1: K+2/+3 side
  const int m    = lane & 15;
  const int col  = blockIdx.y * 128 + wave * 16 + m;

  v8f acc = {0.f, 0.f, 0.f, 0.f, 0.f, 0.f, 0.f, 0.f};

#pragma unroll 4
  for (int j = 0; j < 64; ++j) {
    const int k0 = 4 * j + 2 * half;
    v2f a;
    a.x = sA[m * 260 + k0];
    a.y = sA[m * 260 + k0 + 1];
    v2f b;
    b.x = W[(size_t)k0 * kF + col];
    b.y = W[(size_t)(k0 + 1) * kF + col];
    acc = wmma_f32_4(a, b, acc);
  }

  const float bv = bias[col];
#pragma unroll
  for (int r = 0; r < 8; ++r)
    Y[(size_t)(row0 + r + 8 * half) * kF + col] = acc[r] + bv;
}

// ---------------------------------------------------------------------------
// Fused edge logits: for 16 edges per wave,
//   ee = eattr_tile(16x64) @ We(64x256) via WMMA (16 col-tiles x 16 K-steps),
//   m  = ee + xl[src] + xr[dst]  (gathers, L2-resident),
//   logit[e,h] = sum_c leakyrelu(m)*att ; atomicMax into lmax[dst,h].
// No LDS; eattr fragments live in registers, We from L2 (64KB, hot).
// ---------------------------------------------------------------------------
__global__ void edge_logits_kernel(const long long* __restrict__ ei,
                                   const float* __restrict__ eattr,
                                   const float* __restrict__ loopAttr,
                                   const float* __restrict__ We,
                                   const float* __restrict__ att,
                                   const float* __restrict__ xl,
                                   const float* __restrict__ xr,
                                   float* __restrict__ logits,
                                   unsigned* __restrict__ lmaxU) {
  const int wave = threadIdx.x >> 5;
  const int lane = threadIdx.x & 31;
  const int g    = blockIdx.x * 8 + wave;
  if (g >= kGroups) return;             // wave-uniform: EXEC stays all-ones
  const int eb   = g * 16;
  const int half = lane >> 4;
  const int m    = lane & 15;

  // Row metadata + A fragments. Lane L and L+16 both own matrix row (L&15).
  const int er = eb + m;
  int s_reg, d_reg;
  const float* rp;
  if (er < kE) {
    s_reg = (int)ei[er];
    d_reg = (int)ei[kE + er];
    rp    = eattr + (size_t)er * kED;
  } else {
    s_reg = er - kE;
    d_reg = er - kE;
    rp    = loopAttr + (size_t)(er - kE) * kED;
  }

  v2f afrag[16];
  const float2* rp2 = (const float2*)rp;
#pragma unroll
  for (int j = 0; j < 16; ++j) {
    float2 t = rp2[2 * j + half];       // K = 4j + 2*half, +1
    v2f a; a.x = t.x; a.y = t.y;
    afrag[j] = a;
  }

  int srows[8], drows[8];
#pragma unroll
  for (int r = 0; r < 8; ++r) {
    const int rid = r + 8 * half;       // D-layout row for VGPR r
    srows[r] = __shfl(s_reg, rid, 32);
    drows[r] = __shfl(d_reg, rid, 32);
  }

  float hacc[8][4] = {};

#pragma unroll
  for (int t = 0; t < 16; ++t) {
    v8f acc = {0.f, 0.f, 0.f, 0.f, 0.f, 0.f, 0.f, 0.f};
    const int col = t * 16 + m;
#pragma unroll
    for (int j = 0; j < 16; ++j) {
      const int krow = 4 * j + 2 * half;
      v2f b;
      b.x = We[(size_t)krow * kF + col];
      b.y = We[(size_t)(krow + 1) * kF + col];
      acc = wmma_f32_4(afrag[j], b, acc);
    }
    const float av = att[col];
    const int   h  = t >> 2;            // 4 tiles per head
#pragma unroll
    for (int r = 0; r < 8; ++r) {
      float mm = acc[r] + xl[(size_t)srows[r] * kF + col]
                        + xr[(size_t)drows[r] * kF + col];
      float lr = fmaxf(mm, 0.2f * mm);  // leaky_relu, slope 0.2
      hacc[r][h] += lr * av;
    }
  }

  // Cross-lane reduce the 16 column-owners; lanes m==0 hold the sums.
#pragma unroll
  for (int r = 0; r < 8; ++r) {
#pragma unroll
    for (int h = 0; h < 4; ++h) {
      float v = hacc[r][h];
      v += __shfl_xor(v, 1, 32);
      v += __shfl_xor(v, 2, 32);
      v += __shfl_xor(v, 4, 32);
      v += __shfl_xor(v, 8, 32);
      if (m == 0) {
        const int e = eb + r + 8 * half;
        logits[(size_t)e * kHeads + h] = v;
        atomicMax(&lmaxU[(size_t)drows[r] * kHeads + h], f2ord(v));
      }
    }
  }
}

// ---------------------------------------------------------------------------
// a = exp(logit - lmax[dst]);  denom[dst] += a  (stored in-place over logits)
// ---------------------------------------------------------------------------
__global__ void edge_exp_kernel(const long long* __restrict__ ei,
                                float* __restrict__ logits,
                                const unsigned* __restrict__ lmaxU,
                                float* __restrict__ denom) {
  int idx = blockIdx.x * 256 + threadIdx.x;
  if (idx >= kEf * kHeads) return;
  const int e = idx >> 2;
  const int h = idx & 3;
  const int d = (e < kE) ? (int)ei[kE + e] : (e - kE);
  const float mx = ord2f(lmaxU[(size_t)d * kHeads + h]);
  const float a  = __expf(logits[idx] - mx);
  logits[idx] = a;
  atomicAdd(&denom[(size_t)d * kHeads + h], a);
}

// ---------------------------------------------------------------------------
// out[dst] += alpha * xl[src]   (one wave per edge, 8 channels per lane,
// fp32 global atomics resolved in L2 — out is 51MB, L2-resident)
// ---------------------------------------------------------------------------
__global__ void aggregate_kernel(const long long* __restrict__ ei,
                                 const float* __restrict__ num,
                                 const float* __restrict__ denom,
                                 const float* __restrict__ xl,
                                 float* __restrict__ xout) {
  const int wave = threadIdx.x >> 5;
  const int lane = threadIdx.x & 31;
  const long long e = (long long)blockIdx.x * 8 + wave;
  if (e >= kEf) return;
  int s, d;
  if (e < kE) { s = (int)ei[e]; d = (int)ei[kE + e]; }
  else        { s = d = (int)(e - kE); }

  float alpha[4];
#pragma unroll
  for (int h = 0; h < 4; ++h)
    alpha[h] = num[(size_t)e * kHeads + h] / denom[(size_t)d * kHeads + h];

  const float* xs = xl   + (size_t)s * kF;
  float*       xo = xout + (size_t)d * kF;
#pragma unroll
  for (int j = 0; j < 8; ++j) {
    const int c = j * 32 + lane;
    atomicAdd(&xo[c], alpha[c >> 6] * xs[c]);
  }
}

// ---------------------------------------------------------------------------
// Driver
// ---------------------------------------------------------------------------
extern "C" void kernel_launch(void* const* d_in, const int* in_sizes, int n_in,
                              void* d_out, int out_size, void* d_ws, size_t ws_size,
                              hipStream_t stream) {
  const float*     x     = (const float*)d_in[0];
  const long long* ei    = (const long long*)d_in[1];   // int64 [2,E]
  const float*     eattr = (const float*)d_in[2];
  const float*     Wl    = (const float*)d_in[3];
  const float*     bl    = (const float*)d_in[4];
  const float*     Wr    = (const float*)d_in[5];
  const float*     br    = (const float*)d_in[6];
  const float*     We    = (const float*)d_in[7];
  const float*     att   = (const float*)d_in[8];
  const float*     bias  = (const float*)d_in[9];
  (void)in_sizes; (void)n_in; (void)out_size; (void)ws_size;

  // Workspace layout (floats). Total ~233 MB.
  float* ws       = (float*)d_ws;
  float* xl       = ws;                                  // N*256
  float* xr       = xl + (size_t)kN * kF;                // N*256
  float* bufA     = xr + (size_t)kN * kF;                // N*256
  float* bufB     = bufA + (size_t)kN * kF;              // N*256
  float* loopAttr = bufB + (size_t)kN * kF;              // N*64 (sum -> mean)
  float* cnt      = loopAttr + (size_t)kN * kED;         // N
  float* logitsB  = cnt + kN;                            // Ef*4
  unsigned* lmaxU = (unsigned*)(logitsB + (size_t)kEf * kHeads);  // N*4
  float* denom    = (float*)(lmaxU + (size_t)kN * kHeads);        // N*4

  // Zero loop_attr accumulator + counts (adjacent).
  hipMemsetAsync(loopAttr, 0, ((size_t)kN * kED + kN) * sizeof(float), stream);

  pre_sum_kernel<<<(kE * kED + 255) / 256, 256, 0, stream>>>(eattr, ei, loopAttr, cnt);
  pre_div_kernel<<<(kN * kED + 255) / 256, 256, 0, stream>>>(loopAttr, cnt);

  for (int l = 0; l < kLayers; ++l) {
    const float* WlL = Wl + (size_t)l * kF * kF;
    const float* blL = bl + (size_t)l * kF;
    const float* WrL = Wr + (size_t)l * kF * kF;
    const float* brL = br + (size_t)l * kF;
    const float* WeL = We + (size_t)l * kED * kF;
    const float* atL = att + (size_t)l * kF;    // [4][64] flat
    const float* biL = bias + (size_t)l * kF;

    const float* xin  = (l == 0) ? x : ((l == 1) ? bufA : bufB);
    float*       xout = (l == 0) ? bufA : ((l == 1) ? bufB : (float*)d_out);

    init_nh_kernel<<<(kN * kHeads + 255) / 256, 256, 0, stream>>>(lmaxU, denom);
    init_out_kernel<<<(kN * kF + 255) / 256, 256, 0, stream>>>(xout, biL);

    node_gemm_kernel<<<dim3(kN / 16, 2, 2), 256, 0, stream>>>(
        xin, WlL, blL, WrL, brL, xl, xr);

    edge_logits_kernel<<<(kGroups + 7) / 8, 256, 0, stream>>>(
        ei, eattr, loopAttr, WeL, atL, xl, xr, logitsB, lmaxU);

    edge_exp_kernel<<<(kEf * kHeads + 255) / 256, 256, 0, stream>>>(
        ei, logitsB, lmaxU, denom);

    aggregate_kernel<<<(kEf + 7) / 8, 256, 0, stream>>>(
        ei, logitsB, denom, xl, xout);
  }
}